// LiLTSelfAttention_29557964931081
// MI455X (gfx1250) — compile-verified
//
#include <hip/hip_runtime.h>

// LiLT dual-stream self-attention, fp32, CDNA5 (gfx1250) WMMA F32 16x16x4.
// Forward pass: stop_gradient is identity => probs_text == probs_layout,
// so a single online softmax feeds both ctx (P@V) and lctx (P@LV).

typedef __attribute__((ext_vector_type(2))) float v2f;
typedef __attribute__((ext_vector_type(8))) float v8f;

#define WMMA_F32(a, b, c) \
  __builtin_amdgcn_wmma_f32_16x16x4_f32(false, (a), false, (b), (short)0, (c), false, false)

static constexpr int B_   = 4;
static constexpr int S_   = 2048;
static constexpr int D_   = 768;
static constexpr int LD_  = 192;
static constexpr int H_   = 12;
static constexpr int DH_  = 64;
static constexpr int LDH_ = 16;

// ---------------------------------------------------------------------------
// C = X[M,K] @ W[K,N] + bias, output scattered into [B,H,S,dh] layout.
// Block = 8 waves. The block stages a 32(K) x 64(N) W tile in LDS (shared by
// all waves); each wave computes a 32(M) x 64(N) output tile via fp32 WMMA.
// ---------------------------------------------------------------------------
__global__ __launch_bounds__(256)
void wmma_qkv_gemm(const float* __restrict__ X, const float* __restrict__ W,
                   const float* __restrict__ bias, float* __restrict__ out,
                   int M, int N, int K, int S, int H, int dh)
{
    __shared__ alignas(16) float sB[32 * 68];   // rows padded: bank-conflict free

    const int wave = threadIdx.x >> 5;
    const int lane = threadIdx.x & 31;
    const int half = lane >> 4;
    const int l16  = lane & 15;
    const int m0 = (blockIdx.y * 8 + wave) * 32;
    const int n0 = blockIdx.x * 64;

    const float* xrow0 = X + (size_t)(m0 + l16) * K;
    const float* xrow1 = xrow0 + (size_t)16 * K;

    v8f acc[2][4] = {{{}, {}, {}, {}}, {{}, {}, {}, {}}};

    for (int k0 = 0; k0 < K; k0 += 32) {
        __syncthreads();
        // stage W[k0..k0+32)[n0..n0+64) : 512 float4, 2 per thread
        for (int i = threadIdx.x; i < 512; i += 256) {
            const int row = i >> 4, c4 = i & 15;
            *(float4*)(sB + row * 68 + c4 * 4) =
                *(const float4*)(W + (size_t)(k0 + row) * N + n0 + c4 * 4);
        }
        if (k0 + 32 < K)
            __builtin_prefetch(W + (size_t)(k0 + 32 + (threadIdx.x >> 4)) * N +
                                   n0 + (threadIdx.x & 15) * 4, 0, 3);
        __syncthreads();

#pragma unroll
        for (int kk = 0; kk < 32; kk += 4) {
            const int ka = kk + 2 * half;
            const v2f a0 = *(const v2f*)(xrow0 + k0 + ka);
            const v2f a1 = *(const v2f*)(xrow1 + k0 + ka);
#pragma unroll
            for (int t = 0; t < 4; ++t) {
                v2f b;
                b[0] = sB[ka * 68 + t * 16 + l16];
                b[1] = sB[(ka + 1) * 68 + t * 16 + l16];
                acc[0][t] = WMMA_F32(a0, b, acc[0][t]);
                acc[1][t] = WMMA_F32(a1, b, acc[1][t]);
            }
        }
    }

#pragma unroll
    for (int rb = 0; rb < 2; ++rb) {
#pragma unroll
        for (int t = 0; t < 4; ++t) {
            const int n = n0 + t * 16 + l16;
            const float bv = bias[n];
            const int h = n / dh;
            const int d = n % dh;
#pragma unroll
            for (int r = 0; r < 8; ++r) {
                const int m  = m0 + rb * 16 + r + 8 * half;
                const int bb = m / S;
                const int s  = m % S;
                out[(((size_t)bb * H + h) * S + s) * dh + d] = acc[rb][t][r] + bv;
            }
        }
    }
}

// ---------------------------------------------------------------------------
// Fused dual-stream flash attention. Block = 8 waves; each wave owns a
// 16-query tile; waves share K/V/LK/LV key tiles through LDS (padded rows).
// ---------------------------------------------------------------------------
__global__ __launch_bounds__(256)
void lilt_attention(const float* __restrict__ q,  const float* __restrict__ k,
                    const float* __restrict__ v,  const float* __restrict__ lq,
                    const float* __restrict__ lk, const float* __restrict__ lv,
                    const float* __restrict__ mask,
                    float* __restrict__ ctx, float* __restrict__ lctx)
{
    __shared__ alignas(16) float sK[16 * 68];
    __shared__ alignas(16) float sV[16 * 68];
    __shared__ alignas(16) float sLK[16 * 20];
    __shared__ alignas(16) float sLV[16 * 20];
    __shared__ alignas(16) float sP[8][16 * 18];   // per-wave P transpose buffer

    const int bh   = blockIdx.y;
    const int b    = bh / H_;
    const int h    = bh % H_;
    const int wave = threadIdx.x >> 5;
    const int lane = threadIdx.x & 31;
    const int half = lane >> 4;
    const int l16  = lane & 15;
    const int q0   = blockIdx.x * 128 + wave * 16;

    const float* Qb  = q  + (size_t)bh * S_ * DH_;
    const float* Kb  = k  + (size_t)bh * S_ * DH_;
    const float* Vb  = v  + (size_t)bh * S_ * DH_;
    const float* LQb = lq + (size_t)bh * S_ * LDH_;
    const float* LKb = lk + (size_t)bh * S_ * LDH_;
    const float* LVb = lv + (size_t)bh * S_ * LDH_;
    const float* Mb  = mask + (size_t)b * S_;
    float* sPw = &sP[wave][0];

    // Q fragments, pre-scaled by 1/sqrt(DH); LQ by 1/sqrt(LDH).
    v2f qa[16], lqa[4];
#pragma unroll
    for (int c = 0; c < 16; ++c) {
        const int ka = 4 * c + 2 * half;
        v2f t = *(const v2f*)(Qb + (size_t)(q0 + l16) * DH_ + ka);
        qa[c][0] = t[0] * 0.125f;
        qa[c][1] = t[1] * 0.125f;
    }
#pragma unroll
    for (int c = 0; c < 4; ++c) {
        const int ka = 4 * c + 2 * half;
        v2f t = *(const v2f*)(LQb + (size_t)(q0 + l16) * LDH_ + ka);
        lqa[c][0] = t[0] * 0.25f;
        lqa[c][1] = t[1] * 0.25f;
    }

    v8f ot[4] = {{}, {}, {}, {}};
    v8f ol = {};
    float mrun[8], lrun[8];
#pragma unroll
    for (int r = 0; r < 8; ++r) { mrun[r] = -3.0e38f; lrun[r] = 0.0f; }

    for (int kt = 0; kt < S_ / 16; ++kt) {
        const int key0 = kt * 16;
        __syncthreads();
        {
            const int row = threadIdx.x >> 4, c4 = threadIdx.x & 15;
            *(float4*)(sK + row * 68 + c4 * 4) =
                *(const float4*)(Kb + (size_t)(key0 + row) * 64 + c4 * 4);
            *(float4*)(sV + row * 68 + c4 * 4) =
                *(const float4*)(Vb + (size_t)(key0 + row) * 64 + c4 * 4);
        }
        if (threadIdx.x < 64) {
            const int row = threadIdx.x >> 2, c4 = threadIdx.x & 3;
            *(float4*)(sLK + row * 20 + c4 * 4) =
                *(const float4*)(LKb + (size_t)(key0 + row) * 16 + c4 * 4);
        } else if (threadIdx.x < 128) {
            const int j = threadIdx.x - 64;
            const int row = j >> 2, c4 = j & 3;
            *(float4*)(sLV + row * 20 + c4 * 4) =
                *(const float4*)(LVb + (size_t)(key0 + row) * 16 + c4 * 4);
        }
        if (kt + 1 < S_ / 16) {   // prefetch next tile (global_prefetch_b8)
            __builtin_prefetch(Kb + (size_t)(key0 + 16) * 64 + threadIdx.x * 4, 0, 3);
            __builtin_prefetch(Vb + (size_t)(key0 + 16) * 64 + threadIdx.x * 4, 0, 3);
        }
        __syncthreads();

        // scores tile = Q·Kt/8 + LQ·LKt/4   (16x16 f32 accumulator)
        v8f c = {};
#pragma unroll
        for (int cc = 0; cc < 16; ++cc) {
            const int ka = 4 * cc + 2 * half;
            const v2f bf = *(const v2f*)(sK + l16 * 68 + ka);
            c = WMMA_F32(qa[cc], bf, c);
        }
#pragma unroll
        for (int cc = 0; cc < 4; ++cc) {
            const int ka = 4 * cc + 2 * half;
            const v2f bf = *(const v2f*)(sLK + l16 * 20 + ka);
            c = WMMA_F32(lqa[cc], bf, c);
        }
        const float mk = Mb[key0 + l16];

        // online softmax: each row lives in one 16-lane half at fixed VGPR r
#pragma unroll
        for (int r = 0; r < 8; ++r) {
            float x = c[r] + mk;
            float rowmax = x;
            rowmax = fmaxf(rowmax, __shfl_xor(rowmax, 1));
            rowmax = fmaxf(rowmax, __shfl_xor(rowmax, 2));
            rowmax = fmaxf(rowmax, __shfl_xor(rowmax, 4));
            rowmax = fmaxf(rowmax, __shfl_xor(rowmax, 8));
            const float mnew = fmaxf(mrun[r], rowmax);
            const float sc   = __expf(mrun[r] - mnew);
            const float p    = __expf(x - mnew);
            float rs = p;
            rs += __shfl_xor(rs, 1);
            rs += __shfl_xor(rs, 2);
            rs += __shfl_xor(rs, 4);
            rs += __shfl_xor(rs, 8);
            lrun[r] = lrun[r] * sc + rs;
            mrun[r] = mnew;
            ot[0][r] *= sc; ot[1][r] *= sc; ot[2][r] *= sc; ot[3][r] *= sc;
            ol[r] *= sc;
            // stash P in C/D layout -> read back in A layout (wave-private)
            sPw[(r + 8 * half) * 18 + l16] = p;
        }
        asm volatile("s_wait_dscnt 0" ::: "memory");

        // O += P @ V  and  OL += P @ LV
#pragma unroll
        for (int cc = 0; cc < 4; ++cc) {
            const int ka = 4 * cc + 2 * half;
            const v2f pa = *(const v2f*)(sPw + l16 * 18 + ka);
#pragma unroll
            for (int t = 0; t < 4; ++t) {
                v2f bf;
                bf[0] = sV[ka * 68 + t * 16 + l16];
                bf[1] = sV[(ka + 1) * 68 + t * 16 + l16];
                ot[t] = WMMA_F32(pa, bf, ot[t]);
            }
            v2f bf;
            bf[0] = sLV[ka * 20 + l16];
            bf[1] = sLV[(ka + 1) * 20 + l16];
            ol = WMMA_F32(pa, bf, ol);
        }
    }

#pragma unroll
    for (int r = 0; r < 8; ++r) {
        const int s = q0 + r + 8 * half;
        const float inv = 1.0f / lrun[r];
#pragma unroll
        for (int t = 0; t < 4; ++t) {
            ctx[((size_t)b * S_ + s) * (H_ * DH_) + h * DH_ + t * 16 + l16] =
                ot[t][r] * inv;
        }
        lctx[((size_t)b * S_ + s) * (H_ * LDH_) + h * LDH_ + l16] = ol[r] * inv;
    }
}

extern "C" void kernel_launch(void* const* d_in, const int* in_sizes, int n_in,
                              void* d_out, int out_size, void* d_ws, size_t ws_size,
                              hipStream_t stream)
{
    (void)in_sizes; (void)n_in; (void)out_size; (void)ws_size;
    const float* hs   = (const float*)d_in[0];
    const float* lhs  = (const float*)d_in[1];
    const float* mask = (const float*)d_in[2];
    const float* Wq  = (const float*)d_in[3];
    const float* bq  = (const float*)d_in[4];
    const float* Wk  = (const float*)d_in[5];
    const float* bk  = (const float*)d_in[6];
    const float* Wv  = (const float*)d_in[7];
    const float* bv  = (const float*)d_in[8];
    const float* LWq = (const float*)d_in[9];
    const float* Lbq = (const float*)d_in[10];
    const float* LWk = (const float*)d_in[11];
    const float* Lbk = (const float*)d_in[12];
    const float* LWv = (const float*)d_in[13];
    const float* Lbv = (const float*)d_in[14];

    const size_t n_t = (size_t)B_ * H_ * S_ * DH_;   // 6291456
    const size_t n_l = (size_t)B_ * H_ * S_ * LDH_;  // 1572864
    float* qd  = (float*)d_ws;
    float* kd  = qd + n_t;
    float* vd  = kd + n_t;
    float* lqd = vd + n_t;
    float* lkd = lqd + n_l;
    float* lvd = lkd + n_l;

    const int M = B_ * S_;
    dim3 blk(256);
    dim3 gT(D_ / 64, M / 256);
    wmma_qkv_gemm<<<gT, blk, 0, stream>>>(hs, Wq, bq, qd, M, D_, D_, S_, H_, DH_);
    wmma_qkv_gemm<<<gT, blk, 0, stream>>>(hs, Wk, bk, kd, M, D_, D_, S_, H_, DH_);
    wmma_qkv_gemm<<<gT, blk, 0, stream>>>(hs, Wv, bv, vd, M, D_, D_, S_, H_, DH_);
    dim3 gL(LD_ / 64, M / 256);
    wmma_qkv_gemm<<<gL, blk, 0, stream>>>(lhs, LWq, Lbq, lqd, M, LD_, LD_, S_, H_, LDH_);
    wmma_qkv_gemm<<<gL, blk, 0, stream>>>(lhs, LWk, Lbk, lkd, M, LD_, LD_, S_, H_, LDH_);
    wmma_qkv_gemm<<<gL, blk, 0, stream>>>(lhs, LWv, Lbv, lvd, M, LD_, LD_, S_, H_, LDH_);

    float* ctx  = (float*)d_out;
    float* lctx = ctx + (size_t)B_ * S_ * H_ * DH_;
    dim3 ga(S_ / 128, B_ * H_);
    lilt_attention<<<ga, dim3(256), 0, stream>>>(qd, kd, vd, lqd, lkd, lvd, mask, ctx, lctx);
}